// GATconv_28776280883393
// MI455X (gfx1250) — compile-verified
//
#include <hip/hip_runtime.h>
#include <math.h>

typedef __attribute__((ext_vector_type(2))) float v2f;
typedef __attribute__((ext_vector_type(8))) float v8f;

#define N_NODES   50000
#define N_EDGES   1600000
#define IN_F      128
#define HEADS     4
#define OUT_F     32
#define NOUT      (HEADS * OUT_F)   /* 128 */
#define NEG_SLOPE 0.2f

// ---------------------------------------------------------------------------
// monotone float <-> uint encoding for atomicMax on floats
// ---------------------------------------------------------------------------
__device__ __forceinline__ unsigned enc_f32(float f) {
    unsigned u = __float_as_uint(f);
    return (u & 0x80000000u) ? ~u : (u | 0x80000000u);
}
__device__ __forceinline__ float dec_f32(unsigned e) {
    return (e & 0x80000000u) ? __uint_as_float(e ^ 0x80000000u)
                             : __uint_as_float(~e);
}
#define ENC_NEG_INF 0x007FFFFFu   /* enc(-inf) */

__device__ __forceinline__ void atomic_add_f32(float* p, float v) {
    __hip_atomic_fetch_add(p, v, __ATOMIC_RELAXED, __HIP_MEMORY_SCOPE_AGENT);
}

// ---------------------------------------------------------------------------
// Kernel 1: h = x @ W  via V_WMMA_F32_16X16X4_F32 (wave32)
//   block = 256 threads (8 waves). Block owns a 16-row stripe of x (staged in
//   LDS, row stride padded to 132 words). Wave w owns cols [16w, 16w+16).
//   A layout (16x4 f32):  lane<16: M=lane,   a.{x,y}=K{0,1};  lane>=16: K{2,3}
//   B layout (4x16 f32):  lane<16: N=lane,   b.{x,y}=K{0,1};  lane>=16: K{2,3}
//   C/D layout: acc[r] = D[r + 8*(lane>>4)][lane&15]
// ---------------------------------------------------------------------------
__global__ __launch_bounds__(256) void gemm_wmma(const float* __restrict__ x,
                                                 const float* __restrict__ W,
                                                 float* __restrict__ h) {
    __shared__ float lds_x[16 * 132];
    const int row0 = blockIdx.x * 16;
    const int tid  = threadIdx.x;

    // stage 16x128 tile of x, coalesced 32B per lane
    {
        const int r = tid >> 4;
        const int c = (tid & 15) * 8;
        const float4* src = (const float4*)(x + (long long)(row0 + r) * IN_F + c);
        float4 a = src[0];
        float4 b = src[1];
        *(float4*)(&lds_x[r * 132 + c])     = a;
        *(float4*)(&lds_x[r * 132 + c + 4]) = b;
    }
    __syncthreads();

    const int wave = tid >> 5;
    const int lane = tid & 31;
    const int half = lane >> 4;
    const int sub  = lane & 15;
    const int colb = wave * 16;

    v8f acc = {};
#pragma unroll 8
    for (int k0 = 0; k0 < IN_F; k0 += 4) {
        const int ka = k0 + 2 * half;
        v2f a, b;
        a.x = lds_x[sub * 132 + ka];
        a.y = lds_x[sub * 132 + ka + 1];
        b.x = W[ka * NOUT + colb + sub];
        b.y = W[(ka + 1) * NOUT + colb + sub];
        acc = __builtin_amdgcn_wmma_f32_16x16x4_f32(
            /*neg_a=*/false, a, /*neg_b=*/false, b,
            /*c_mod=*/(short)0, acc, /*reuse_a=*/false, /*reuse_b=*/false);
    }
#pragma unroll
    for (int r = 0; r < 8; ++r) {
        h[(long long)(row0 + half * 8 + r) * NOUT + colb + sub] = acc[r];
    }
}

// ---------------------------------------------------------------------------
// Kernel 2: e_src[n,h] = <h[n,h,:], att_src[h,:]>, e_dst likewise.
//   block = 128 threads = 4 waves; wave w = head w; lane = feature.
// ---------------------------------------------------------------------------
__global__ __launch_bounds__(128) void attn_logits(const float* __restrict__ h,
                                                   const float* __restrict__ att,
                                                   float* __restrict__ e_src,
                                                   float* __restrict__ e_dst) {
    const int n    = blockIdx.x;
    const int w    = threadIdx.x >> 5;
    const int lane = threadIdx.x & 31;
    const float v  = h[(long long)n * NOUT + w * OUT_F + lane];
    float es = v * att[w * (2 * OUT_F) + lane];
    float ed = v * att[w * (2 * OUT_F) + OUT_F + lane];
#pragma unroll
    for (int o = 16; o; o >>= 1) {
        es += __shfl_xor(es, o, 32);
        ed += __shfl_xor(ed, o, 32);
    }
    if (lane == 0) {
        e_src[n * HEADS + w] = es;
        e_dst[n * HEADS + w] = ed;
    }
}

// ---------------------------------------------------------------------------
// Kernel 3: out = bias (broadcast), amax = enc(-inf), denom = 0
// ---------------------------------------------------------------------------
__global__ __launch_bounds__(256) void init_buffers(float* __restrict__ out,
                                                    const float* __restrict__ bias,
                                                    unsigned* __restrict__ amax_bits,
                                                    float* __restrict__ denom) {
    const long long idx = (long long)blockIdx.x * blockDim.x + threadIdx.x;
    if (idx < (long long)N_NODES * NOUT) out[idx] = bias[idx & (NOUT - 1)];
    if (idx < (long long)N_NODES * HEADS) {
        amax_bits[idx] = ENC_NEG_INF;
        denom[idx]     = 0.0f;
    }
}

// ---------------------------------------------------------------------------
// Kernel 4: segment max of leaky-relu(e_src[row]+e_dst[col]) over rows
// ---------------------------------------------------------------------------
__global__ __launch_bounds__(256) void edge_max(const int* __restrict__ row,
                                                const int* __restrict__ col,
                                                const float* __restrict__ e_src,
                                                const float* __restrict__ e_dst,
                                                unsigned* __restrict__ amax_bits) {
    const long long t = (long long)blockIdx.x * blockDim.x + threadIdx.x;
    if (t >= (long long)N_EDGES * HEADS) return;
    const int e  = (int)(t >> 2);
    const int hd = (int)(t & 3);
    const int r  = row[e];
    const int c  = col[e];
    float a = e_src[r * HEADS + hd] + e_dst[c * HEADS + hd];
    a = (a >= 0.0f) ? a : NEG_SLOPE * a;
    atomicMax(&amax_bits[r * HEADS + hd], enc_f32(a));
}

// ---------------------------------------------------------------------------
// Kernel 5: ex = exp(a - amax[row]); stash per-edge; denom[row] += ex
// ---------------------------------------------------------------------------
__global__ __launch_bounds__(256) void edge_expsum(const int* __restrict__ row,
                                                   const int* __restrict__ col,
                                                   const float* __restrict__ e_src,
                                                   const float* __restrict__ e_dst,
                                                   const unsigned* __restrict__ amax_bits,
                                                   float* __restrict__ exw,
                                                   float* __restrict__ denom) {
    const long long t = (long long)blockIdx.x * blockDim.x + threadIdx.x;
    if (t >= (long long)N_EDGES * HEADS) return;
    const int e  = (int)(t >> 2);
    const int hd = (int)(t & 3);
    const int r  = row[e];
    const int c  = col[e];
    float a = e_src[r * HEADS + hd] + e_dst[c * HEADS + hd];
    a = (a >= 0.0f) ? a : NEG_SLOPE * a;
    const float m  = dec_f32(amax_bits[r * HEADS + hd]);
    const float ex = __expf(a - m);
    exw[t] = ex;
    atomic_add_f32(&denom[r * HEADS + hd], ex);
}

// ---------------------------------------------------------------------------
// Kernel 6: out[row, h, :] += (ex/denom) * h[col, h, :]
//   one wave per (edge, head); lane = feature (coalesced 128B gather)
// ---------------------------------------------------------------------------
__global__ __launch_bounds__(256) void aggregate(const int* __restrict__ row,
                                                 const int* __restrict__ col,
                                                 const float* __restrict__ exw,
                                                 const float* __restrict__ denom,
                                                 const float* __restrict__ h,
                                                 float* __restrict__ out) {
    const long long t   = (long long)blockIdx.x * blockDim.x + threadIdx.x;
    const long long wid = t >> 5;
    if (wid >= (long long)N_EDGES * HEADS) return;
    const int lane = (int)(t & 31);
    const int e    = (int)(wid >> 2);
    const int hd   = (int)(wid & 3);
    const int r    = row[e];
    const int c    = col[e];
    const float coef = exw[wid] / (denom[r * HEADS + hd] + 1e-16f);
    const float val  = coef * h[(long long)c * NOUT + hd * OUT_F + lane];
    atomic_add_f32(&out[(long long)r * NOUT + hd * OUT_F + lane], val);
}

// ---------------------------------------------------------------------------
// Host-side launcher
// inputs: 0:x (N*128 f32), 1:indices (2*E int32), 2:weight (128*128 f32),
//         3:att (4*64 f32), 4:bias (128 f32)
// ---------------------------------------------------------------------------
extern "C" void kernel_launch(void* const* d_in, const int* in_sizes, int n_in,
                              void* d_out, int out_size, void* d_ws, size_t ws_size,
                              hipStream_t stream) {
    const float* x    = (const float*)d_in[0];
    const int*   idx  = (const int*)d_in[1];
    const float* W    = (const float*)d_in[2];
    const float* att  = (const float*)d_in[3];
    const float* bias = (const float*)d_in[4];
    float* out = (float*)d_out;

    const int* row = idx;
    const int* col = idx + N_EDGES;

    // workspace layout (all 256B-aligned)
    char* ws = (char*)d_ws;
    size_t off = 0;
    auto take = [&](size_t bytes) {
        char* p = ws + off;
        off = (off + bytes + 255) & ~(size_t)255;
        return p;
    };
    float*    h_buf  = (float*)take((size_t)N_NODES * NOUT * sizeof(float));
    float*    e_src  = (float*)take((size_t)N_NODES * HEADS * sizeof(float));
    float*    e_dst  = (float*)take((size_t)N_NODES * HEADS * sizeof(float));
    unsigned* amax_b = (unsigned*)take((size_t)N_NODES * HEADS * sizeof(unsigned));
    float*    denom  = (float*)take((size_t)N_NODES * HEADS * sizeof(float));
    float*    exw    = (float*)take((size_t)N_EDGES * HEADS * sizeof(float));
    (void)ws_size;

    // 1) h = x @ W (WMMA)
    gemm_wmma<<<N_NODES / 16, 256, 0, stream>>>(x, W, h_buf);

    // 2) attention logits per node
    attn_logits<<<N_NODES, 128, 0, stream>>>(h_buf, att, e_src, e_dst);

    // 3) init out/amax/denom
    {
        long long n = (long long)N_NODES * NOUT;
        init_buffers<<<(unsigned)((n + 255) / 256), 256, 0, stream>>>(out, bias, amax_b, denom);
    }

    // 4) segment max
    {
        long long n = (long long)N_EDGES * HEADS;
        edge_max<<<(unsigned)((n + 255) / 256), 256, 0, stream>>>(row, col, e_src, e_dst, amax_b);
    }

    // 5) exp + segment sum
    {
        long long n = (long long)N_EDGES * HEADS;
        edge_expsum<<<(unsigned)((n + 255) / 256), 256, 0, stream>>>(row, col, e_src, e_dst,
                                                                     amax_b, exw, denom);
    }

    // 6) weighted aggregation
    {
        long long n = (long long)N_EDGES * HEADS * 32;
        aggregate<<<(unsigned)((n + 255) / 256), 256, 0, stream>>>(row, col, exw, denom, h_buf, out);
    }
}